// Hypernym_44014824849512
// MI455X (gfx1250) — compile-verified
//
#include <hip/hip_runtime.h>

// ---------------------------------------------------------------------------
// Weighted embedding-bag for MI455X (gfx1250).
//   out[b, :] = sum_l weights[b,l] * emb_table[ids[b,l], :]
//   B=16384, L=20, V=100000, D=300  (all fp32)
//
// Memory-bound (0.5 FLOP/byte, ~416 MB total traffic). The 120-MB table fits
// the 192-MB L2 and is reused ~3.3x -> keep it resident; stream the 19.7-MB
// output with non-temporal stores. WMMA does not apply (per-batch gathered
// matrices -> no shared GEMM operand); the CDNA5 feature that matters is the
// Tensor Data Mover *gather mode*: DMA 20 scattered 1200-byte embedding rows
// into LDS per batch row (TENSORcnt), then reduce from LDS with ds_load_b128.
// ---------------------------------------------------------------------------

#define B_SZ 16384
#define L_SZ 20
#define V_SZ 100000
#define D_SZ 300

typedef unsigned int       u32;
typedef unsigned long long u64;
typedef u32   __attribute__((ext_vector_type(4))) u32x4;
typedef int   __attribute__((ext_vector_type(4))) i32x4;
typedef int   __attribute__((ext_vector_type(8))) i32x8;
typedef float __attribute__((ext_vector_type(4))) f32x4;

__global__ __launch_bounds__(64) void hypernym_bag_kernel(
    const int*   __restrict__ ids,      // [B, L]
    const float* __restrict__ weights,  // [B, L]
    const float* __restrict__ emb,      // [V, D]
    float*       __restrict__ out)      // [B, D]
{
#if defined(__HIP_DEVICE_COMPILE__)
  // 20 rows x 300 floats = 24000 bytes, gathered here by the TDM.
  __shared__ __align__(16) float smem[L_SZ * D_SZ];

  const int b   = blockIdx.x;
  const int tid = threadIdx.x;

  // ---------------- Phase 1: TDM gather (wave 0 issues) ----------------
  if (tid < 32) {
    const int* idrow = ids + b * L_SZ;
    int idx[L_SZ];
#pragma unroll
    for (int l = 0; l < L_SZ; ++l)
      idx[l] = __builtin_amdgcn_readfirstlane(idrow[l]);   // uniform per block

    const u64 base = (u64)(uintptr_t)emb;

    // Three gather-mode descriptors: rows 0-7, 8-15, 16-19 (32-bit indices,
    // max 8 rows per descriptor). Each gathers n rows of 300 f32 into LDS.
#pragma unroll
    for (int c = 0; c < 3; ++c) {
      const int n       = (c == 2) ? 4 : 8;         // valid indices this chunk
      const u32 lds_off = (u32)(c * 8 * D_SZ * 4);  // rows packed consecutively

      // D# group 0: count=1 | gather_index_size=32b | gather_mode=1,
      //             lds_addr, global_addr (57b), type=2 ("image")
      u32x4 g0;
      g0.x = 1u | (1u << 30) | (1u << 31);
      g0.y = lds_off;
      g0.z = (u32)(base & 0xffffffffull);
      g0.w = (u32)((base >> 32) & 0x1ffffffull) | (2u << 30);

      // D# group 1: data_size=4B; tensor_dim0=300 (row length);
      //             tensor_dim1=100000 (row count, OOB check);
      //             tile_dim0=300; tile_dim1=#indices; dim0_stride=300
      i32x8 g1;
      g1[0] = (int)(2u << 16);                                  // data_size=2 (4B)
      g1[1] = (int)(((u32)D_SZ & 0xffffu) << 16);               // tensor_dim0 lo16
      g1[2] = (int)(((u32)V_SZ & 0xffffu) << 16);               // dim0 hi=0 | dim1 lo16
      g1[3] = (int)(((u32)V_SZ >> 16) | ((u32)D_SZ << 16));     // dim1 hi | tile_dim0
      g1[4] = n;                                                // tile_dim1 = #indices
      g1[5] = D_SZ;                                             // dim0_stride lo32
      g1[6] = 0;
      g1[7] = 0;

      // D# groups 2/3: 32-bit row indices (8 per descriptor).
      const int b0 = c * 8;
      i32x4 g2, g3;
      g2[0] = idx[b0 + 0]; g2[1] = idx[b0 + 1];
      g2[2] = idx[b0 + 2]; g2[3] = idx[b0 + 3];
      if (c == 2) {
        g3[0] = idx[19]; g3[1] = idx[19]; g3[2] = idx[19]; g3[3] = idx[19];
      } else {
        g3[0] = idx[b0 + 4]; g3[1] = idx[b0 + 5];
        g3[2] = idx[b0 + 6]; g3[3] = idx[b0 + 7];
      }

      // amdgpu-toolchain (clang-23) 6-arg form:
      //   (v4u32, v8i32, v4i32, v4i32, v8i32, i32 cpol)
      i32x8 g4 = {0, 0, 0, 0, 0, 0, 0, 0};
      __builtin_amdgcn_tensor_load_to_lds(g0, g1, g2, g3, g4, /*cpol=*/0);
    }
    __builtin_amdgcn_s_wait_tensorcnt(0);   // all 3 DMA gathers complete
  }
  __syncthreads();

  // Escape the LDS address into an opaque asm with a memory clobber so the
  // compiler cannot assume the TDM never wrote smem (no visible C++ stores).
  asm volatile("" ::"v"(&smem[0]) : "memory");

  // ---------------- Phase 2: weighted reduction from LDS ----------------
  float w[L_SZ];
  const float* wrow = weights + b * L_SZ;
#pragma unroll
  for (int l = 0; l < L_SZ; ++l) w[l] = wrow[l];

  const f32x4* lds4 = (const f32x4*)smem;              // 75 f32x4 per row
  f32x4*       out4 = (f32x4*)(out + (size_t)b * D_SZ);

  for (int c = tid; c < D_SZ / 4; c += 64) {
    f32x4 acc = {0.f, 0.f, 0.f, 0.f};
#pragma unroll
    for (int l = 0; l < L_SZ; ++l) {
      const f32x4 v = lds4[l * (D_SZ / 4) + c];        // ds_load_b128
      acc.x = fmaf(w[l], v.x, acc.x);
      acc.y = fmaf(w[l], v.y, acc.y);
      acc.z = fmaf(w[l], v.z, acc.z);
      acc.w = fmaf(w[l], v.w, acc.w);
    }
    // Streaming output: non-temporal store keeps the emb table L2-resident.
    __builtin_nontemporal_store(acc, &out4[c]);        // global_store_b128 NT
  }
#endif
}

extern "C" void kernel_launch(void* const* d_in, const int* in_sizes, int n_in,
                              void* d_out, int out_size, void* d_ws, size_t ws_size,
                              hipStream_t stream) {
  (void)in_sizes; (void)n_in; (void)out_size; (void)d_ws; (void)ws_size;
  const int*   ids     = (const int*)d_in[0];
  const float* weights = (const float*)d_in[1];
  const float* emb     = (const float*)d_in[2];
  float*       out     = (float*)d_out;

  hypernym_bag_kernel<<<dim3(B_SZ), dim3(64), 0, stream>>>(ids, weights, emb, out);
}